// GATGNN_43379169689806
// MI455X (gfx1250) — compile-verified
//
#include <hip/hip_runtime.h>
#include <math.h>

#define D 64
#define NEG_SLOPE 0.2f

typedef float v2f __attribute__((ext_vector_type(2)));
typedef float v8f __attribute__((ext_vector_type(8)));

// ---- monotone float <-> ordered-int mapping for atomic float max ----
__device__ __forceinline__ int f2o(float f) {
    int i = __float_as_int(f);
    return (i >= 0) ? i : (i ^ 0x7FFFFFFF);
}
__device__ __forceinline__ float o2f(int i) {
    return __int_as_float((i >= 0) ? i : (i ^ 0x7FFFFFFF));
}
#define ORDERED_NEG_INF 0x807FFFFF  // f2o(-inf)

// -------------------------------------------------------------------------
// Init per layer: out = bias (broadcast over rows), smax = -inf, denom = 0
// -------------------------------------------------------------------------
__global__ void layer_init(float* __restrict__ out, const float* __restrict__ bias,
                           int* __restrict__ smax_o, float* __restrict__ denom, int n_nodes) {
    int i = blockIdx.x * blockDim.x + threadIdx.x;
    int total = n_nodes * D;
    if (i < total) out[i] = bias[i & (D - 1)];
    if (i < n_nodes) {
        smax_o[i] = ORDERED_NEG_INF;
        denom[i] = 0.0f;
    }
}

// -------------------------------------------------------------------------
// h = maybe_relu(in) @ W  via V_WMMA_F32_16X16X4_F32.
// One wave computes a full 16x64 output slab: 4 accumulators share each A
// operand (4x fewer A loads, 4 independent WMMA dep-chains per k-step).
// grid = (ceil(N/16)), block = 32.
// A (16x4 f32): lanes 0-15 M=0..15 {v0:K=0, v1:K=1}; lanes 16-31 {v0:K=2, v1:K=3}
// B (4x16 f32): lanes 0-15 N=0..15 {v0:K=0, v1:K=1}; lanes 16-31 {v0:K=2, v1:K=3}
// C/D (16x16 f32): vgpr r -> M = r + 8*(lane>>4), N = lane&15
// -------------------------------------------------------------------------
template <bool RELU>
__global__ __launch_bounds__(32) void gemm_wmma_f32(const float* __restrict__ in,
                                                    const float* __restrict__ W,
                                                    float* __restrict__ h,
                                                    int n_nodes) {
    const int tm   = blockIdx.x;
    const int lane = threadIdx.x & 31;
    const int half = lane >> 4;   // 0/1
    const int l15  = lane & 15;

    int row   = tm * 16 + l15;                     // A row (M)
    int row_c = row < n_nodes ? row : n_nodes - 1; // clamp loads, keep EXEC full

    v8f acc0 = {}, acc1 = {}, acc2 = {}, acc3 = {};
#pragma unroll
    for (int k0 = 0; k0 < D; k0 += 4) {
        const int ka = k0 + 2 * half;
        float a0 = in[(size_t)row_c * D + ka];
        float a1 = in[(size_t)row_c * D + ka + 1];
        if (RELU) { a0 = fmaxf(a0, 0.0f); a1 = fmaxf(a1, 0.0f); }
        v2f a; a[0] = a0; a[1] = a1;

        const float* Wr0 = W + (size_t)ka * D;
        const float* Wr1 = W + (size_t)(ka + 1) * D;
        v2f b0; b0[0] = Wr0[l15     ]; b0[1] = Wr1[l15     ];
        v2f b1; b1[0] = Wr0[l15 + 16]; b1[1] = Wr1[l15 + 16];
        v2f b2; b2[0] = Wr0[l15 + 32]; b2[1] = Wr1[l15 + 32];
        v2f b3; b3[0] = Wr0[l15 + 48]; b3[1] = Wr1[l15 + 48];

        acc0 = __builtin_amdgcn_wmma_f32_16x16x4_f32(false, a, false, b0, (short)0, acc0, false, false);
        acc1 = __builtin_amdgcn_wmma_f32_16x16x4_f32(false, a, false, b1, (short)0, acc1, false, false);
        acc2 = __builtin_amdgcn_wmma_f32_16x16x4_f32(false, a, false, b2, (short)0, acc2, false, false);
        acc3 = __builtin_amdgcn_wmma_f32_16x16x4_f32(false, a, false, b3, (short)0, acc3, false, false);
    }
#pragma unroll
    for (int r = 0; r < 8; ++r) {
        int m = tm * 16 + r + 8 * half;
        if (m < n_nodes) {
            float* hr = h + (size_t)m * D;
            hr[l15     ] = acc0[r];
            hr[l15 + 16] = acc1[r];
            hr[l15 + 32] = acc2[r];
            hr[l15 + 48] = acc3[r];
        }
    }
}

// -------------------------------------------------------------------------
// Per-node attention dots: s_src[n] = h[n].a_src ; s_dst[n] = h[n].a_dst
// One wave32 per node, shuffle reduce.
// -------------------------------------------------------------------------
__global__ void node_dots(const float* __restrict__ h,
                          const float* __restrict__ a_src, const float* __restrict__ a_dst,
                          float* __restrict__ s_src, float* __restrict__ s_dst, int n_nodes) {
    int gid  = blockIdx.x * blockDim.x + threadIdx.x;
    int node = gid >> 5;
    int lane = threadIdx.x & 31;
    if (node >= n_nodes) return;
    const float* hr = h + (size_t)node * D;
    float x0 = hr[lane], x1 = hr[lane + 32];
    float ps = x0 * a_src[lane] + x1 * a_src[lane + 32];
    float pd = x0 * a_dst[lane] + x1 * a_dst[lane + 32];
#pragma unroll
    for (int off = 16; off > 0; off >>= 1) {
        ps += __shfl_down(ps, off, 32);
        pd += __shfl_down(pd, off, 32);
    }
    if (lane == 0) { s_src[node] = ps; s_dst[node] = pd; }
}

__device__ __forceinline__ void edge_endpoints(const int* __restrict__ ei, int e,
                                               int n_edges_raw, int& s, int& d) {
    if (e < n_edges_raw) { s = ei[e]; d = ei[n_edges_raw + e]; }
    else                 { s = d = e - n_edges_raw; }   // self-loop
}

// -------------------------------------------------------------------------
// Per-edge score = leaky_relu(s_src[src] + s_dst[dst]); segment max over dst
// -------------------------------------------------------------------------
__global__ void edge_score_max(const int* __restrict__ ei,
                               const float* __restrict__ s_src, const float* __restrict__ s_dst,
                               float* __restrict__ score, int* __restrict__ smax_o,
                               int n_edges_raw, int n_nodes) {
    int e = blockIdx.x * blockDim.x + threadIdx.x;
    int total = n_edges_raw + n_nodes;
    if (e >= total) return;
    int s, d;
    edge_endpoints(ei, e, n_edges_raw, s, d);
    float sc = s_src[s] + s_dst[d];
    sc = (sc > 0.0f) ? sc : NEG_SLOPE * sc;
    score[e] = sc;
    atomicMax(&smax_o[d], f2o(sc));
}

// -------------------------------------------------------------------------
// ex = exp(score - smax[dst]) (in place over score buf); denom[dst] += ex
// -------------------------------------------------------------------------
__global__ void edge_exp_sum(const int* __restrict__ ei,
                             float* __restrict__ score_ex, const int* __restrict__ smax_o,
                             float* __restrict__ denom, int n_edges_raw, int n_nodes) {
    int e = blockIdx.x * blockDim.x + threadIdx.x;
    int total = n_edges_raw + n_nodes;
    if (e >= total) return;
    int s, d;
    edge_endpoints(ei, e, n_edges_raw, s, d);
    float v = __expf(score_ex[e] - o2f(smax_o[d]));
    score_ex[e] = v;
    atomicAdd(&denom[d], v);
}

// -------------------------------------------------------------------------
// out[dst] += (ex[e]/denom[dst]) * h[src]   (64 threads per edge)
// -------------------------------------------------------------------------
__global__ void edge_scatter(const int* __restrict__ ei,
                             const float* __restrict__ ex, const float* __restrict__ denom,
                             const float* __restrict__ h, float* __restrict__ out,
                             int n_edges_raw, int n_nodes) {
    long long idx = (long long)blockIdx.x * blockDim.x + threadIdx.x;
    int e = (int)(idx >> 6);
    int dim = (int)(idx & (D - 1));
    int total = n_edges_raw + n_nodes;
    if (e >= total) return;
    int s, d;
    edge_endpoints(ei, e, n_edges_raw, s, d);
    float alpha = ex[e] / denom[d];
    atomicAdd(&out[(size_t)d * D + dim], alpha * h[(size_t)s * D + dim]);
}

// -------------------------------------------------------------------------
// Host side
// -------------------------------------------------------------------------
extern "C" void kernel_launch(void* const* d_in, const int* in_sizes, int n_in,
                              void* d_out, int out_size, void* d_ws, size_t ws_size,
                              hipStream_t stream) {
    const float* x  = (const float*)d_in[1];
    const int*   ei = (const int*)d_in[2];
    const int n_nodes = in_sizes[1] / D;     // 50000
    const int n_edges = in_sizes[2] / 2;     // 800000
    const int e_total = n_edges + n_nodes;   // + self loops

    // workspace carve-up (floats)
    float* ws    = (float*)d_ws;
    float* hbuf  = ws;                              // N*D
    float* io0   = hbuf + (size_t)n_nodes * D;      // N*D
    float* s_src = io0  + (size_t)n_nodes * D;      // N
    float* s_dst = s_src + n_nodes;                 // N
    int*   smax  = (int*)(s_dst + n_nodes);         // N
    float* denom = (float*)(smax + n_nodes);        // N
    float* exbuf = denom + n_nodes;                 // E+N

    float* outf = (float*)d_out;

    const int tiles_m      = (n_nodes + 15) / 16;
    const int init_blocks  = (n_nodes * D + 255) / 256;
    const int dot_blocks   = (n_nodes * 32 + 255) / 256;           // wave per node
    const int edge_blocks  = (e_total + 255) / 256;
    const long long sc_thr = (long long)e_total * D;
    const int scat_blocks  = (int)((sc_thr + 255) / 256);

    struct Layer { const float* in; float* out; int relu_in; int wi; };
    Layer layers[3] = {
        { x,    outf, 0, 3  },   // W1,a_src1,a_dst1,b1 at d_in[3..6]
        { outf, io0,  1, 7  },
        { io0,  outf, 1, 11 },
    };

    for (int li = 0; li < 3; ++li) {
        const float* W     = (const float*)d_in[layers[li].wi + 0];
        const float* a_src = (const float*)d_in[layers[li].wi + 1];
        const float* a_dst = (const float*)d_in[layers[li].wi + 2];
        const float* bias  = (const float*)d_in[layers[li].wi + 3];

        layer_init<<<init_blocks, 256, 0, stream>>>(layers[li].out, bias, smax, denom, n_nodes);
        if (layers[li].relu_in)
            gemm_wmma_f32<true ><<<tiles_m, 32, 0, stream>>>(layers[li].in, W, hbuf, n_nodes);
        else
            gemm_wmma_f32<false><<<tiles_m, 32, 0, stream>>>(layers[li].in, W, hbuf, n_nodes);
        node_dots<<<dot_blocks, 256, 0, stream>>>(hbuf, a_src, a_dst, s_src, s_dst, n_nodes);
        edge_score_max<<<edge_blocks, 256, 0, stream>>>(ei, s_src, s_dst, exbuf, smax,
                                                        n_edges, n_nodes);
        edge_exp_sum<<<edge_blocks, 256, 0, stream>>>(ei, exbuf, smax, denom,
                                                      n_edges, n_nodes);
        edge_scatter<<<scat_blocks, 256, 0, stream>>>(ei, exbuf, denom, hbuf, layers[li].out,
                                                      n_edges, n_nodes);
    }
}